// AdvancedQuantumLayer_13288628814300
// MI455X (gfx1250) — compile-verified
//
#include <hip/hip_runtime.h>
#include <math.h>

typedef __attribute__((ext_vector_type(2))) float v2f;
typedef __attribute__((ext_vector_type(8))) float v8f;

#define NQ 12
#define NL 4
#define NSTATE 4096
#define QBATCH 256

// ---------------------------------------------------------------------------
// Kernel 1: fuse each (layer, group-of-4-qubits) Rot stack into a 16x16
// complex unitary:  fused[j][k] = prod_t U_{q0+t}[bit_{3-t}(j)][bit_{3-t}(k)]
// with U = RZ(om) RY(th) RZ(phi)  (PennyLane Rot).
// 12 blocks (layer*3+group) x 256 threads (one per matrix entry).
// Uses hardware v_sin/v_cos (__sinf/__cosf) -- args are O(pi)*N(0,1).
// ---------------------------------------------------------------------------
__global__ void qfuse_gates_kernel(const float* __restrict__ W,
                                   float* __restrict__ gR,
                                   float* __restrict__ gI) {
  const int id = blockIdx.x;            // 0..11
  const int L = id / 3, g = id % 3, q0 = g * 4;
  const int t = threadIdx.x;            // 0..255  ->  (j,k) entry
  const int j = t >> 4, k = t & 15;
  float re = 1.0f, im = 0.0f;
#pragma unroll
  for (int tt = 0; tt < 4; ++tt) {
    const int q = q0 + tt;
    const float phi = W[(L * NQ + q) * 3 + 0];
    const float th  = W[(L * NQ + q) * 3 + 1];
    const float om  = W[(L * NQ + q) * 3 + 2];
    const float c  = __cosf(th * 0.5f), s = __sinf(th * 0.5f);
    const float ap = (phi + om) * 0.5f;   // (phi+omega)/2
    const float am = (phi - om) * 0.5f;   // (phi-omega)/2
    const float ca = __cosf(ap), sa = __sinf(ap);
    const float cb = __cosf(am), sb = __sinf(am);
    const int jb = (j >> (3 - tt)) & 1;
    const int kb = (k >> (3 - tt)) & 1;
    // U00 = ( ca*c, -sa*c)   U01 = (-cb*s, -sb*s)
    // U10 = ( cb*s, -sb*s)   U11 = ( ca*c,  sa*c)
    const bool diag = (jb == kb);
    const float mag = diag ? c : s;                      // |entry| factor
    float ur = (diag ? ca : cb) * mag;
    if (!diag && kb == 1) ur = -ur;                      // U01 sign
    float ui = (diag ? sa : sb) * mag;
    if (!(jb == 1 && kb == 1)) ui = -ui;                 // only U11 has +sa
    const float nr = re * ur - im * ui;
    im = re * ui + im * ur;
    re = nr;
  }
  gR[id * 256 + t] = re;
  gI[id * 256 + t] = im;
}

// element index for group G (compile-time), nibble value k, column c
template <int G>
__device__ __forceinline__ int gidx_t(int k, int c) {
  if (G == 0) return (k << 8) | c;
  if (G == 1) return ((c & 0xF0) << 4) | (k << 4) | (c & 15);
  return (c << 4) | k;
}

// Apply one fused 16x16 complex gate to the LDS state via FP32 WMMA.
// 16 column tiles of 16; 8 waves -> 2 tiles/wave (fully unrolled).
// Tiles are disjoint per wave and wave LDS ops are in-order, so in-place safe.
template <int G>
__device__ __forceinline__ void apply_gate(const float* __restrict__ ur,
                                           const float* __restrict__ ui,
                                           float* __restrict__ sR,
                                           float* __restrict__ sI,
                                           int wave, int hi, int m) {
  // A(16x4) fragment per K-step: lane m holds row m;
  // v0 = K(base + 2*hi), v1 = K(base + 2*hi + 1)
  v2f aR[4], aI[4], aIn[4];
#pragma unroll
  for (int s4 = 0; s4 < 4; ++s4) {
    const int kk = s4 * 4 + hi * 2;
    aR[s4][0]  = ur[m * 16 + kk];   aR[s4][1]  = ur[m * 16 + kk + 1];
    aI[s4][0]  = ui[m * 16 + kk];   aI[s4][1]  = ui[m * 16 + kk + 1];
    aIn[s4][0] = -aI[s4][0];        aIn[s4][1] = -aI[s4][1];
  }
#pragma unroll
  for (int tt = 0; tt < 2; ++tt) {
    const int c = (wave + tt * 8) * 16 + m;   // this lane's column
    v8f cr = {}; v8f ci = {};
#pragma unroll
    for (int s4 = 0; s4 < 4; ++s4) {
      const int kk = s4 * 4 + hi * 2;
      const int i0 = gidx_t<G>(kk, c);
      const int i1 = gidx_t<G>(kk + 1, c);
      v2f br, bi;
      br[0] = sR[i0]; br[1] = sR[i1];
      bi[0] = sI[i0]; bi[1] = sI[i1];
      // D_r += Ar*Br - Ai*Bi ; D_i += Ar*Bi + Ai*Br
      cr = __builtin_amdgcn_wmma_f32_16x16x4_f32(false, aR[s4],  false, br, (short)0, cr, false, false);
      cr = __builtin_amdgcn_wmma_f32_16x16x4_f32(false, aIn[s4], false, bi, (short)0, cr, false, false);
      ci = __builtin_amdgcn_wmma_f32_16x16x4_f32(false, aR[s4],  false, bi, (short)0, ci, false, false);
      ci = __builtin_amdgcn_wmma_f32_16x16x4_f32(false, aI[s4],  false, br, (short)0, ci, false, false);
    }
    // D layout: vgpr r, lane -> row r + 8*hi, col m
#pragma unroll
    for (int r = 0; r < 8; ++r) {
      const int o = gidx_t<G>(r + 8 * hi, c);
      sR[o] = cr[r]; sI[o] = ci[r];
    }
  }
}

// ---------------------------------------------------------------------------
// Kernel 2: one block = one batch sample. 32KB state lives in LDS.
// ---------------------------------------------------------------------------
__global__ __launch_bounds__(256) void qsim_kernel(
    const float* __restrict__ X,       // (256,12)
    const float* __restrict__ ENT,     // (4,12)
    const float* __restrict__ gR,      // (12,256) fused gate real
    const float* __restrict__ gI,      // (12,256) fused gate imag
    float* __restrict__ OUT) {         // (256,12)
  __shared__ float sR[NSTATE];
  __shared__ float sI[NSTATE];
  __shared__ float eR[NQ][2];
  __shared__ float eI[NQ][2];
  __shared__ float zsum[NQ];

  const int b = blockIdx.x;
  const int t = threadIdx.x;
  const int lane = t & 31;
  const int wave = t >> 5;
  const int hi = lane >> 4;      // lane half
  const int m  = lane & 15;

  const float PI_F = 3.14159265358979323846f;

  if (t < NQ) {
    const float x = X[b * NQ + t];
    const float y = x * PI_F;          // RY angle
    const float z = x * x * PI_F;      // RZ angle
    const float cy = __cosf(y * 0.5f), sy = __sinf(y * 0.5f);
    const float cz = __cosf(z * 0.5f), sz = __sinf(z * 0.5f);
    // a(0) = e^{-i z/2} cos(y/2),  a(1) = e^{+i z/2} sin(y/2)
    eR[t][0] =  cz * cy; eI[t][0] = -sz * cy;
    eR[t][1] =  cz * sy; eI[t][1] =  sz * sy;
    zsum[t] = 0.0f;
  }
  __syncthreads();

  // encoded state is a product state: psi(s) = prod_q a_q(bit_q(s))
#pragma unroll 1
  for (int u = 0; u < 16; ++u) {
    const int s = t * 16 + u;
    float re = 1.0f, im = 0.0f;
#pragma unroll
    for (int q = 0; q < NQ; ++q) {
      const int bit = (s >> (11 - q)) & 1;
      const float ar = eR[q][bit], ai = eI[q][bit];
      const float nr = re * ar - im * ai;
      im = re * ai + im * ar;
      re = nr;
    }
    sR[s] = re; sI[s] = im;
  }
  __syncthreads();

#pragma unroll 1
  for (int L = 0; L < NL; ++L) {
    const float* baR = gR + L * 3 * 256;
    const float* baI = gI + L * 3 * 256;
    apply_gate<0>(baR,       baI,       sR, sI, wave, hi, m); __syncthreads();
    apply_gate<1>(baR + 256, baI + 256, sR, sI, wave, hi, m); __syncthreads();
    apply_gate<2>(baR + 512, baI + 512, sR, sI, wave, hi, m); __syncthreads();

    // block-uniform 12-bit CNOT mask, forced into an SGPR for scalar branches
    unsigned lm = 0u;
#pragma unroll
    for (int qq = 0; qq < NQ; ++qq)
      lm |= (ENT[L * NQ + qq] > 0.5f) ? (1u << qq) : 0u;
    lm = (unsigned)__builtin_amdgcn_readfirstlane((int)lm);

    // masked ring CNOTs, reference order q = 0..11; each pass = 1024 disjoint
    // swaps (4/thread), indices built branch-free with constant bit-inserts.
#pragma unroll
    for (int q = 0; q < NQ; ++q) {
      if (lm & (1u << q)) {
        const int cbit = 11 - q;
        const int tbit = 11 - ((q + 1) % NQ);
        const int lo  = (cbit < tbit) ? cbit : tbit;
        const int hi2 = (cbit < tbit) ? tbit : cbit;
#pragma unroll
        for (int pp = 0; pp < 4; ++pp) {
          unsigned s = (unsigned)(t * 4 + pp);          // 10 free bits
          s = ((s & ~((1u << lo)  - 1u)) << 1) | (s & ((1u << lo)  - 1u));
          s = ((s & ~((1u << hi2) - 1u)) << 1) | (s & ((1u << hi2) - 1u));
          s |= 1u << cbit;                               // control = 1
          const unsigned s2 = s | (1u << tbit);          // target partner
          const float r0 = sR[s],  i0 = sI[s];
          const float r1 = sR[s2], i1 = sI[s2];
          sR[s]  = r1; sI[s]  = i1;
          sR[s2] = r0; sI[s2] = i0;
        }
      }
      __syncthreads();
    }
  }

  // ---- <Z_q> = sum_s |psi(s)|^2 * (1 - 2*bit_q(s)) ----
  float acc[NQ];
#pragma unroll
  for (int q = 0; q < NQ; ++q) acc[q] = 0.0f;
#pragma unroll 1
  for (int u = 0; u < 16; ++u) {
    const int s = t * 16 + u;
    const float pr = sR[s] * sR[s] + sI[s] * sI[s];
#pragma unroll
    for (int q = 0; q < NQ; ++q)
      acc[q] += ((s >> (11 - q)) & 1) ? -pr : pr;
  }
#pragma unroll
  for (int q = 0; q < NQ; ++q) atomicAdd(&zsum[q], acc[q]);
  __syncthreads();
  if (t < NQ) OUT[b * NQ + t] = zsum[t];
}

extern "C" void kernel_launch(void* const* d_in, const int* in_sizes, int n_in,
                              void* d_out, int out_size, void* d_ws, size_t ws_size,
                              hipStream_t stream) {
  (void)in_sizes; (void)n_in; (void)out_size; (void)ws_size;
  const float* x   = (const float*)d_in[0];  // (256,12)
  const float* w   = (const float*)d_in[1];  // (4,12,3)
  const float* ent = (const float*)d_in[2];  // (4,12)
  float* out = (float*)d_out;                // (256,12)
  float* gR = (float*)d_ws;                  // 12*256 floats
  float* gI = gR + 12 * 256;                 // 12*256 floats  (24KB total)
  qfuse_gates_kernel<<<12, 256, 0, stream>>>(w, gR, gI);
  qsim_kernel<<<QBATCH, 256, 0, stream>>>(x, ent, gR, gI, out);
}